// RoiPoolingConv_73443940762096
// MI455X (gfx1250) — compile-verified
//
#include <hip/hip_runtime.h>
#include <hip/hip_bf16.h>

// RoiPoolingConv for MI455X (gfx1250).
// img:  [1, 200, 200, 512] f32 (NHWC, C innermost, stride-1)
// rois: [1, 300, 4] i32 (x, y, w, h)
// out:  [1, 300, 7, 7, 512] f32
//
// Memory-bound gather (~0.4 FLOP/byte, roofline floor ~6.5us at 23.3 TB/s).
// Feature map (81.9 MB) fits in 192 MB L2 -> direct wide loads capture all
// inter-cell reuse. One block per (ROI, py) pooled row: ROI descriptor load,
// the f32 divisions and y-side math amortize over 7 px cells, and each thread
// carries 28 independent B128 loads for deep memory ILP.

typedef float v4f __attribute__((ext_vector_type(4)));

#define IMG_H 200
#define IMG_W 200
#define IMG_C 512
#define POOL  7
#define NROIS 300

__global__ __launch_bounds__(128) void roi_pool_bilinear_kernel(
    const float* __restrict__ img,
    const int*   __restrict__ rois,
    float*       __restrict__ out)
{
    // blockIdx.x in [0, NROIS*POOL): one block per pooled output ROW (r, py).
    const int blk = blockIdx.x;
    const int r   = blk / POOL;
    const int py  = blk - r * POOL;

    // Uniform ROI descriptor -> scalar s_load_b128.
    const int4 roi = *reinterpret_cast<const int4*>(rois + r * 4);
    const int x = roi.x, y = roi.y, w = roi.z, h = roi.w;

    // ---- y-side (shared by all 7 px cells) -------------------------------
    const float sy = (float)y + (float)py * ((float)h / (float)POOL);
    const float fy = floorf(sy);
    int y0 = (int)fy;
    int y1 = min(y0 + 1, y + h - 1);          // reference: from UNclipped y0
    y0 = min(max(y0, 0), IMG_H - 1);
    y1 = min(max(y1, 0), IMG_H - 1);
    const float wy  = sy - fy;                 // weight from unclipped floor
    const float wyc = 1.0f - wy;

    // Each thread owns 4 consecutive channels -> one float4 per corner.
    const int c = (int)threadIdx.x * 4;        // 0..508 (blockDim.x == 128)

    const float* row0 = img + ((long)y0 * IMG_W) * IMG_C + c;
    const float* row1 = img + ((long)y1 * IMG_W) * IMG_C + c;
    float*       orow = out + ((long)blk * POOL) * IMG_C + c;   // (r*7+py)*7 cells

    const float wdiv = (float)w / (float)POOL; // one true f32 div, reused 7x
    const float xf   = (float)x;
    const int   xmax = x + w - 1;

    // Kick the memory system on the first cell while the ALU chain runs.
    {
        const float sx0 = xf;                  // px = 0 -> sx == x exactly
        int x0 = min(max((int)sx0, 0), IMG_W - 1);
        int x1 = min(max(min((int)sx0 + 1, xmax), 0), IMG_W - 1);
        __builtin_prefetch(row0 + (long)x0 * IMG_C, 0, 1);
        __builtin_prefetch(row0 + (long)x1 * IMG_C, 0, 1);
        __builtin_prefetch(row1 + (long)x0 * IMG_C, 0, 1);
        __builtin_prefetch(row1 + (long)x1 * IMG_C, 0, 1);
    }

#pragma unroll
    for (int px = 0; px < POOL; ++px) {
        const float sx = xf + (float)px * wdiv;
        const float fx = floorf(sx);
        int x0 = (int)fx;
        int x1 = min(x0 + 1, xmax);           // reference: from UNclipped x0
        x0 = min(max(x0, 0), IMG_W - 1);
        x1 = min(max(x1, 0), IMG_W - 1);
        const float wx  = sx - fx;
        const float wxc = 1.0f - wx;

        const v4f v00 = *reinterpret_cast<const v4f*>(row0 + (long)x0 * IMG_C);
        const v4f v01 = *reinterpret_cast<const v4f*>(row0 + (long)x1 * IMG_C);
        const v4f v10 = *reinterpret_cast<const v4f*>(row1 + (long)x0 * IMG_C);
        const v4f v11 = *reinterpret_cast<const v4f*>(row1 + (long)x1 * IMG_C);

        // Exact reference arithmetic form: a*(1-w) + b*w.
        const v4f top = v00 * wxc + v01 * wx;
        const v4f bot = v10 * wxc + v11 * wx;
        const v4f res = top * wyc + bot * wy;

        // Streaming 30 MB output: non-temporal B128 store so it doesn't evict
        // the L2-resident feature map.
        __builtin_nontemporal_store(res,
            reinterpret_cast<v4f*>(orow + (long)px * IMG_C));
    }
}

extern "C" void kernel_launch(void* const* d_in, const int* in_sizes, int n_in,
                              void* d_out, int out_size, void* d_ws, size_t ws_size,
                              hipStream_t stream)
{
    (void)in_sizes; (void)n_in; (void)d_ws; (void)ws_size; (void)out_size;

    const float* img  = reinterpret_cast<const float*>(d_in[0]);
    const int*   rois = reinterpret_cast<const int*>(d_in[1]);
    // d_in[2] is pool_size (== 7, compile-time constant here).
    float* out = reinterpret_cast<float*>(d_out);

    const int nblocks = NROIS * POOL;          // 2,100 pooled rows
    roi_pool_bilinear_kernel<<<dim3(nblocks), dim3(128), 0, stream>>>(img, rois, out);
}